// Watson_62208306315907
// MI455X (gfx1250) — compile-verified
//
#include <hip/hip_runtime.h>
#include <math.h>

#define WATSON_MAX_ITER 50000

typedef __attribute__((ext_vector_type(2))) float v2f;
typedef __attribute__((ext_vector_type(8))) float v8f;

// ---------------------------------------------------------------------------
// Kernel 1: scalars -> ws[0]=logC, ws[1]=kappa_pos, ws[2]=1/||mu||
// log_kummer: cs_j = prefix(base_j) + j*log(kappa_pos),
//             base_j = log((j-.5)/(j*(j+c-1))), logsumexp over {0} ∪ cs.
// 256-thread block: chunked sums + Hillis-Steele scan + online LSE reduce.
// ---------------------------------------------------------------------------
__global__ __launch_bounds__(256) void watson_prelude(
    const float* __restrict__ mu, const float* __restrict__ kappa,
    float* __restrict__ ws, int P)
{
  __shared__ float sA[256];
  __shared__ float sB[256];
  const int tid = threadIdx.x;

  // ---- ||mu||^2 ----
  float v = 0.f;
  for (int i = tid; i < P; i += 256) { float m = mu[i]; v = fmaf(m, m, v); }
  sA[tid] = v; __syncthreads();
  for (int off = 128; off > 0; off >>= 1) {
    if (tid < off) sA[tid] += sA[tid + off];
    __syncthreads();
  }
  const float norm2 = sA[0];
  __syncthreads();

  // ---- kappa_pos = clip(softplus(beta=20, threshold=1)(kappa), 1e-25) ----
  const float k0 = kappa[0];
  float kp = (k0 * 20.f > 1.f) ? k0 : (log1pf(expf(20.f * k0)) * 0.05f);
  kp = fmaxf(kp, 1e-25f);
  const float lk = logf(kp);
  const float c  = 0.5f * (float)P;

  // ---- log_kummer series, j = 1 .. MAX_ITER-1 ----
  const int total = WATSON_MAX_ITER - 1;          // 49999 terms
  const int chunk = (total + 255) / 256;          // 196
  const int jbeg  = 1 + tid * chunk;
  int jend        = jbeg + chunk;
  if (jend > WATSON_MAX_ITER) jend = WATSON_MAX_ITER;

  float local = 0.f;
  for (int j = jbeg; j < jend; ++j) {
    float fj = (float)j;
    local += logf((fj - 0.5f) / (fj * (fj + (c - 1.f))));
  }

  // inclusive scan of per-thread base sums (ping-pong Hillis-Steele)
  sA[tid] = local; __syncthreads();
  float* pin = sA; float* pout = sB;
  for (int off = 1; off < 256; off <<= 1) {
    pout[tid] = pin[tid] + ((tid >= off) ? pin[tid - off] : 0.f);
    __syncthreads();
    float* t = pin; pin = pout; pout = t;
  }
  const float runBase = pin[tid] - local;   // exclusive prefix
  __syncthreads();

  // online logsumexp over this thread's cs_j
  float m = -INFINITY, s = 0.f;
  float run = runBase;
  for (int j = jbeg; j < jend; ++j) {
    float fj = (float)j;
    run += logf((fj - 0.5f) / (fj * (fj + (c - 1.f))));
    float cs = fmaf(fj, lk, run);
    if (cs > m) { s = s * expf(m - cs) + 1.f; m = cs; }
    else        { s += expf(cs - m); }
  }
  if (tid == 0) {          // fold in the leading log(1) = 0 element
    if (0.f > m) { s = s * expf(m) + 1.f; m = 0.f; }
    else         { s += expf(-m); }
  }

  sA[tid] = m; sB[tid] = s; __syncthreads();
  for (int off = 128; off > 0; off >>= 1) {
    if (tid < off) {
      float m1 = sA[tid],       s1 = sB[tid];
      float m2 = sA[tid + off], s2 = sB[tid + off];
      if (s2 > 0.f) {
        if (s1 <= 0.f)    { sA[tid] = m2; sB[tid] = s2; }
        else if (m2 > m1) { sA[tid] = m2; sB[tid] = s1 * expf(m1 - m2) + s2; }
        else              {               sB[tid] = s1 + s2 * expf(m2 - m1); }
      }
    }
    __syncthreads();
  }

  if (tid == 0) {
    float logkum = sA[0] + logf(sB[0]);
    float logSA  = lgammaf(c) - logf(2.f) - c * logf(3.14159265358979323846f);
    ws[0] = logSA - logkum;          // logC
    ws[1] = kp;                      // kappa_pos
    ws[2] = 1.0f / sqrtf(norm2);     // 1/||mu||
  }
}

// ---------------------------------------------------------------------------
// Kernel 2: out[i] = logC + kappa_pos * ((X[i,:]·mu) * inv_norm)^2
// One wave / 16-row tile; lane = (r = row-in-tile, h = K-half).
// KH>0: compile-time half-row length -> batches of 8 independent b128 loads
// (MLP=8) + v_pk_fma_f32 chains. One v_wmma_f32_16x16x4_f32 per tile merges
// the two half partials per row; full tiles store via 2x global_store_b128.
// ---------------------------------------------------------------------------
template<int KH>
__global__ __launch_bounds__(256) void watson_main(
    const float* __restrict__ X, const float* __restrict__ mu,
    const float* __restrict__ ws, float* __restrict__ out,
    int N, int P)
{
  __shared__ __align__(16) float smu[1024];
  for (int i = threadIdx.x; i < P; i += blockDim.x) smu[i] = mu[i];
  __syncthreads();

  const float logC = ws[0];
  const float kp   = ws[1];
  const float inv  = ws[2];

  const int lane = threadIdx.x & 31;
  const int r    = lane & 15;
  const int h    = lane >> 4;
  const int Kh   = (KH > 0) ? KH : (P >> 1);

  const int wavesPerBlk = blockDim.x >> 5;
  const int wave  = blockIdx.x * wavesPerBlk + (threadIdx.x >> 5);
  const int wstep = gridDim.x * wavesPerBlk;
  const int tiles = (N + 15) >> 4;

  for (int t = wave; t < tiles; t += wstep) {
    const int rowBase = t << 4;
    const bool full = (rowBase + 16 <= N);
    int row = rowBase + r;
    if (row >= N) row = N - 1;                     // ragged tail only
    const float* __restrict__ xr = X + (size_t)row * (size_t)P + (size_t)(h * Kh);
    const float* mr = smu + h * Kh;

    float4 a0 = make_float4(0.f, 0.f, 0.f, 0.f);
    float4 a1 = make_float4(0.f, 0.f, 0.f, 0.f);

    if constexpr (KH > 0) {
#pragma unroll
      for (int kb = 0; kb < KH; kb += 32) {        // 4 batches (KH=128)
        float4 xv[8];
#pragma unroll
        for (int u = 0; u < 8; ++u)                // 8 loads in flight
          xv[u] = *reinterpret_cast<const float4*>(xr + kb + 4 * u);
#pragma unroll
        for (int u = 0; u < 8; ++u) {
          const float4 mv = *reinterpret_cast<const float4*>(mr + kb + 4 * u);
          float4& a = (u & 1) ? a1 : a0;
          a.x = fmaf(xv[u].x, mv.x, a.x);
          a.y = fmaf(xv[u].y, mv.y, a.y);
          a.z = fmaf(xv[u].z, mv.z, a.z);
          a.w = fmaf(xv[u].w, mv.w, a.w);
        }
      }
    } else {
      for (int kk = 0; kk < Kh; kk += 4) {
        const float4 xv = *reinterpret_cast<const float4*>(xr + kk);
        const float4 mv = *reinterpret_cast<const float4*>(mr + kk);
        a0.x = fmaf(xv.x, mv.x, a0.x);
        a0.y = fmaf(xv.y, mv.y, a0.y);
        a0.z = fmaf(xv.z, mv.z, a0.z);
        a0.w = fmaf(xv.w, mv.w, a0.w);
      }
    }
    const float p = ((a0.x + a1.x) + (a0.y + a1.y)) + ((a0.z + a1.z) + (a0.w + a1.w));

    // One f32 WMMA combines half-partials: A vgpr0 = p (lanes 0-15 -> (M=r,K=0),
    // lanes 16-31 -> (M=r,K=2)); B col0 = [1,0,1,0] -> D[m,0] = rowSum(m).
    v2f A; A[0] = p;                    A[1] = 0.f;
    v2f B; B[0] = (r == 0) ? 1.f : 0.f; B[1] = 0.f;
    v8f C = {};
    v8f D = __builtin_amdgcn_wmma_f32_16x16x4_f32(
        false, A, false, B, (short)0, C, false, false);

    if (r == 0) {                        // lane 0: rows 0..7, lane 16: rows 8..15
      float res[8];
#pragma unroll
      for (int i = 0; i < 8; ++i) {
        float pr = D[i] * inv;
        res[i] = fmaf(kp, pr * pr, logC);
      }
      float* op = out + rowBase + (h << 3);
      if (full) {
        *reinterpret_cast<float4*>(op)     = make_float4(res[0], res[1], res[2], res[3]);
        *reinterpret_cast<float4*>(op + 4) = make_float4(res[4], res[5], res[6], res[7]);
      } else {
#pragma unroll
        for (int i = 0; i < 8; ++i) {
          int idx = rowBase + (h << 3) + i;
          if (idx < N) out[idx] = res[i];
        }
      }
    }
  }
}

// ---------------------------------------------------------------------------
extern "C" void kernel_launch(void* const* d_in, const int* in_sizes, int n_in,
                              void* d_out, int out_size, void* d_ws, size_t ws_size,
                              hipStream_t stream) {
  const float* X     = (const float*)d_in[0];
  const float* mu    = (const float*)d_in[1];
  const float* kappa = (const float*)d_in[2];
  float* out = (float*)d_out;
  float* ws  = (float*)d_ws;

  const int P = in_sizes[1];     // 256
  const int N = out_size;        // 400000

  watson_prelude<<<1, 256, 0, stream>>>(mu, kappa, ws, P);

  const int tiles  = (N + 15) / 16;          // 25000
  int blocks = (tiles + 7) / 8;              // 8 waves / block -> exact cover
  if (blocks < 1) blocks = 1;
  if (blocks > 1048576) blocks = 1048576;    // grid-stride covers any excess

  if (P == 256) {
    watson_main<128><<<blocks, 256, 0, stream>>>(X, mu, ws, out, N, P);
  } else {
    watson_main<0><<<blocks, 256, 0, stream>>>(X, mu, ws, out, N, P);
  }
}